// Group_2199023255674
// MI455X (gfx1250) — compile-verified
//
#include <hip/hip_runtime.h>

// Problem constants (match reference)
#define B_  32
#define N_  8192
#define G_  512
#define K_  32

#define CHUNK     1024
#define CHUNKP    (CHUNK + 4)         // row pitch: 8*1028 % 64 == 32 -> halves hit disjoint banks
#define NCHUNK    (N_ / CHUNK)        // 8
#define CAND_PER  (NCHUNK * K_)       // 256 candidates per center

typedef __attribute__((ext_vector_type(2))) float v2f;
typedef __attribute__((ext_vector_type(8))) float v8f;
typedef __attribute__((ext_vector_type(4))) unsigned int u32x4;
typedef __attribute__((ext_vector_type(4))) int          i32x4;
typedef __attribute__((ext_vector_type(8))) int          i32x8;

#define FLT_BIG 3.4e38f

// ---------------------------------------------------------------------------
// Stage nFloats of global data into LDS.
// Preferred path: Tensor Data Mover (tensor_load_to_lds) issued by wave 0,
// D# built per CDNA5 ISA ch.8 (1 descriptor, 4-byte elements, 1-row tile).
// LDS byte offset = low 32 bits of the flat pointer (LDS aperture rule).
// ---------------------------------------------------------------------------
__device__ __forceinline__ void stage_lds(float* dstLds, const float* src,
                                          int nFloats, int tid, int nThreads)
{
#if __has_builtin(__builtin_amdgcn_tensor_load_to_lds)
    if (tid < 32) {
        const unsigned      ldsAddr = (unsigned)(uintptr_t)dstLds;
        const unsigned long gaddr   = (unsigned long)(uintptr_t)src;
        const unsigned      ne      = (unsigned)nFloats;   // elements (4B each)

        u32x4 g0;
        g0[0] = 1u;                                        // count=1 (valid descriptor)
        g0[1] = ldsAddr;                                   // lds_addr (bytes)
        g0[2] = (unsigned)(gaddr & 0xffffffffu);           // global_addr[31:0]
        g0[3] = (unsigned)((gaddr >> 32) & 0x01ffffffu)    // global_addr[56:32]
              | (2u << 30);                                // type = 2 ("image")

        i32x8 g1;
        g1[0] = (int)(2u << 16);                           // data_size = 2 (4 bytes)
        g1[1] = (int)((ne & 0xffffu) << 16);               // tensor_dim0[15:0]
        g1[2] = (int)((ne >> 16) | (1u << 16));            // tensor_dim0[31:16] | tensor_dim1=1
        g1[3] = (int)((ne & 0xffffu) << 16);               // tile_dim0 = ne
        g1[4] = 1;                                         // tile_dim1 = 1, tile_dim2 = 0
        g1[5] = (int)ne;                                   // tensor_dim0_stride[31:0]
        g1[6] = (int)((ne & 0xffffu) << 16);               // stride0 hi=0 | stride1[15:0]
        g1[7] = 0;                                         // stride1[47:16]

        i32x4 g2 = {0, 0, 0, 0};
        i32x4 g3 = {0, 0, 0, 0};
#if __clang_major__ >= 23
        i32x8 g4 = {0, 0, 0, 0, 0, 0, 0, 0};
        __builtin_amdgcn_tensor_load_to_lds(g0, g1, g2, g3, g4, 0);
#else
        __builtin_amdgcn_tensor_load_to_lds(g0, g1, g2, g3, 0);
#endif
        __builtin_amdgcn_s_wait_tensorcnt(0);
    }
#else
    for (int i = tid; i < nFloats; i += nThreads) dstLds[i] = src[i];
#endif
    __syncthreads();
}

// ---------------------------------------------------------------------------
// Kernel 1: Farthest Point Sampling (one block per batch).
// Points TDM-staged to LDS once; each thread caches its 8 points in registers,
// so the 512 serial iterations touch only registers + one uniform LDS read.
// ---------------------------------------------------------------------------
__global__ __launch_bounds__(1024) void fps_kernel(const float* __restrict__ xyz,
                                                   float* __restrict__ centerOut)
{
    const int b = blockIdx.x;
    const int t = threadIdx.x;
    const float* xb = xyz + (size_t)b * N_ * 3;

    __shared__ float sPts[N_ * 3];     // 96 KB interleaved xyz
    __shared__ float warpV[32];
    __shared__ int   warpI[32];
    __shared__ int   sFar;

    stage_lds(sPts, xb, N_ * 3, t, 1024);

    float px[8], py[8], pz[8], md[8];
#pragma unroll
    for (int i = 0; i < 8; ++i) {
        const int p = t + i * 1024;
        px[i] = sPts[p * 3 + 0];
        py[i] = sPts[p * 3 + 1];
        pz[i] = sPts[p * 3 + 2];
        md[i] = 1e10f;
    }

    int far = 0;
    for (int g = 0; g < G_; ++g) {
        // current center (recorded BEFORE update, matching reference scan)
        const float cx = sPts[far * 3 + 0];
        const float cy = sPts[far * 3 + 1];
        const float cz = sPts[far * 3 + 2];
        if (t == 0) {
            float* co = centerOut + ((size_t)b * G_ + g) * 3;
            co[0] = cx; co[1] = cy; co[2] = cz;
        }

        float bestV = -1.0f;
        int   bestI = 0x7fffffff;
#pragma unroll
        for (int i = 0; i < 8; ++i) {
            const float dx = px[i] - cx, dy = py[i] - cy, dz = pz[i] - cz;
            const float d  = dx * dx + dy * dy + dz * dz;
            md[i] = fminf(md[i], d);
            const int p = t + i * 1024;
            if (md[i] > bestV || (md[i] == bestV && p < bestI)) { bestV = md[i]; bestI = p; }
        }
        // wave32 argmax butterfly (tie -> lowest index, like jnp.argmax)
        for (int off = 16; off; off >>= 1) {
            const float ov = __shfl_xor(bestV, off);
            const int   oi = __shfl_xor(bestI, off);
            if (ov > bestV || (ov == bestV && oi < bestI)) { bestV = ov; bestI = oi; }
        }
        const int wid = t >> 5;
        if ((t & 31) == 0) { warpV[wid] = bestV; warpI[wid] = bestI; }
        __syncthreads();
        if (wid == 0) {
            float v = warpV[t & 31];
            int   i2 = warpI[t & 31];
            for (int off = 16; off; off >>= 1) {
                const float ov = __shfl_xor(v, off);
                const int   oi = __shfl_xor(i2, off);
                if (ov > v || (ov == v && oi < i2)) { v = ov; i2 = oi; }
            }
            if (t == 0) sFar = i2;
        }
        __syncthreads();
        far = sFar;
    }
}

// ---------------------------------------------------------------------------
// Kernel 2: distance tiles via V_WMMA_F32_16X16X4_F32 + top-32 selection.
// Grid: B * (G/16) blocks; block = 512 threads = 16 waves.
// Points TDM-staged to LDS; fragments built branchlessly from LDS.
// -2 folded into A; |c|^2 + |p|^2 bias folded into the WMMA C accumulator,
// so D = d2 directly and the epilogue is pure stores.
// ---------------------------------------------------------------------------
__global__ __launch_bounds__(512) void group_kernel(const float* __restrict__ xyz,
                                                    const float* __restrict__ center,
                                                    int* __restrict__ idxOut)
{
    const int b    = blockIdx.x >> 5;   // 32 center-tiles per batch
    const int gt   = blockIdx.x & 31;
    const int t    = threadIdx.x;
    const int lane = t & 31;
    const int w    = t >> 5;            // wave id 0..15 == center within tile
    const float* xb = xyz + (size_t)b * N_ * 3;

    __shared__ float sPts[N_ * 3];          // 96 KB interleaved xyz (TDM target)
    __shared__ float sPP[N_];               // 32 KB point squared norms
    __shared__ float sCC[16];               // center squared norms
    __shared__ float d2buf[16 * CHUNKP];    // ~66 KB: 16 centers x 1024 pts (padded pitch)
    __shared__ float candV[16 * CAND_PER];  // 16 KB
    __shared__ int   candI[16 * CAND_PER];  // 16 KB

    stage_lds(sPts, xb, N_ * 3, t, 512);

    for (int p = t; p < N_; p += 512) {
        const float x = sPts[p * 3 + 0], y = sPts[p * 3 + 1], z = sPts[p * 3 + 2];
        sPP[p] = x * x + y * y + z * z;
    }
    if (t < 16) {
        const float* c = center + ((size_t)b * G_ + gt * 16 + t) * 3;
        sCC[t] = c[0] * c[0] + c[1] * c[1] + c[2] * c[2];
    }
    __syncthreads();

    // A fragment (16x4 f32 ISA layout): lanes 0-15 -> (K0,K1)=(x,y) of row M=lane,
    // lanes 16-31 -> (K2,K3)=(z,0). Scaled by -2 so WMMA yields -2*dot.
    const int half = lane >> 4;
    const int mn   = lane & 15;
    v2f afrag;
    {
        const float* c = center + ((size_t)b * G_ + gt * 16 + mn) * 3;
        const float a0 = half ? c[2] : c[0];
        const float a1 = half ? 0.0f : c[1];
        afrag.x = -2.0f * a0;
        afrag.y = -2.0f * a1;
    }
    // Hoist this lane's 8 center-norms (C/D rows M = half*8 + j)
    float ccr[8];
#pragma unroll
    for (int j = 0; j < 8; ++j) ccr[j] = sCC[half * 8 + j];

    for (int ch = 0; ch < NCHUNK; ++ch) {
        const int pbase = ch * CHUNK;

        // --- compute phase: 4 WMMA tiles (16 points each) per wave ---
        for (int tile = 0; tile < 4; ++tile) {
            const int lbase = w * 64 + tile * 16;    // local base within chunk
            const int p     = pbase + lbase + mn;    // this lane's point (column N=mn)
            const int base3 = p * 3;

            // branchless B fragment from LDS: (x,y) or (z,0)
            const float v0 = sPts[base3 + (half << 1)];
            const float v1 = sPts[base3 + 1];
            v2f bfrag;
            bfrag.x = v0;
            bfrag.y = half ? 0.0f : v1;

            // Fold the bias into C: C[m][n] = |c_m|^2 + |p_n|^2 (matches C/D layout:
            // this lane's column is N=mn for all 8 VGPRs, rows are half*8+j).
            const float pp = sPP[p];
            v8f cacc;
#pragma unroll
            for (int j = 0; j < 8; ++j) cacc[j] = ccr[j] + pp;

            cacc = __builtin_amdgcn_wmma_f32_16x16x4_f32(
                /*neg_a=*/false, afrag, /*neg_b=*/false, bfrag,
                /*c_mod=*/(short)0, cacc, /*reuse_a=*/false, /*reuse_b=*/false);

#pragma unroll
            for (int j = 0; j < 8; ++j) {
                const int m = j + half * 8;          // C layout: VGPR j -> rows M / M+8
                d2buf[m * CHUNKP + lbase + mn] = cacc[j];
            }
        }
        __syncthreads();

        // --- selection phase: wave w -> 32 smallest of its 1024-entry row ---
        float* row = d2buf + w * CHUNKP;
        for (int k = 0; k < K_; ++k) {
            float bv = FLT_BIG; int bi = 0x7fffffff;
            for (int j = lane; j < CHUNK; j += 32) {
                const float v = row[j];
                if (v < bv || (v == bv && j < bi)) { bv = v; bi = j; }
            }
            for (int off = 16; off; off >>= 1) {
                const float ov = __shfl_xor(bv, off);
                const int   oi = __shfl_xor(bi, off);
                if (ov < bv || (ov == bv && oi < bi)) { bv = ov; bi = oi; }
            }
            if (lane == 0) {
                row[bi] = FLT_BIG;   // remove winner
                candV[w * CAND_PER + ch * K_ + k] = bv;
                candI[w * CAND_PER + ch * K_ + k] = pbase + bi;
            }
            __builtin_amdgcn_wave_barrier();  // keep DS store ordered before next pass
        }
        __syncthreads();  // d2buf reused next chunk
    }

    // --- final merge: 32 smallest of 256 candidates, ascending order ---
    float* cv = candV + w * CAND_PER;
    int*   ci = candI + w * CAND_PER;
    int*   out = idxOut + ((size_t)b * G_ + gt * 16 + w) * (size_t)K_;
    for (int k = 0; k < K_; ++k) {
        float bv = FLT_BIG; int bi = 0x7fffffff;
        for (int j = lane; j < CAND_PER; j += 32) {
            const float v = cv[j];
            if (v < bv || (v == bv && j < bi)) { bv = v; bi = j; }
        }
        for (int off = 16; off; off >>= 1) {
            const float ov = __shfl_xor(bv, off);
            const int   oi = __shfl_xor(bi, off);
            if (ov < bv || (ov == bv && oi < bi)) { bv = ov; bi = oi; }
        }
        if (lane == 0) {
            cv[bi] = FLT_BIG;
            out[k] = ci[bi];
        }
        __builtin_amdgcn_wave_barrier();
    }
}

// ---------------------------------------------------------------------------
extern "C" void kernel_launch(void* const* d_in, const int* in_sizes, int n_in,
                              void* d_out, int out_size, void* d_ws, size_t ws_size,
                              hipStream_t stream)
{
    (void)in_sizes; (void)n_in; (void)out_size; (void)d_ws; (void)ws_size;

    const float* xyz = (const float*)d_in[0];

    // d_out layout (flat, return order): int32 idx[B*G*K] then float center[B*G*3]
    int*   idxOut    = (int*)d_out;
    float* centerOut = reinterpret_cast<float*>(idxOut + (size_t)B_ * G_ * K_);

    fps_kernel<<<B_, 1024, 0, stream>>>(xyz, centerOut);
    group_kernel<<<B_ * (G_ / 16), 512, 0, stream>>>(xyz, centerOut, idxOut);
}